// GCN_2_16896401342682
// MI455X (gfx1250) — compile-verified
//
#include <hip/hip_runtime.h>
#include <hip/hip_bf16.h>

typedef __attribute__((ext_vector_type(2))) float v2f;
typedef __attribute__((ext_vector_type(8))) float v8f;

#define NODE_D 64
#define NEG_SLOPE 0.01f

// ---------------------------------------------------------------------------
// Degree accumulation: deg[dst] += 1 per edge (self-loop +1 added later)
// ---------------------------------------------------------------------------
__global__ void gcn_deg_count(const int* __restrict__ dst, int E,
                              float* __restrict__ deg) {
    int e = blockIdx.x * blockDim.x + threadIdx.x;
    if (e < E) atomicAdd(&deg[dst[e]], 1.0f);
}

// deg -> dinv_sqrt (in place) and invdeg (separate buffer)
__global__ void gcn_deg_finalize(float* __restrict__ deg_io,
                                 float* __restrict__ invdeg, int n) {
    int i = blockIdx.x * blockDim.x + threadIdx.x;
    if (i >= n) return;
    float d = deg_io[i] + 1.0f;           // + self loop
    deg_io[i] = __frsqrt_rn(d);           // D^-1/2
    invdeg[i] = 1.0f / d;                 // self-loop weight
}

// ---------------------------------------------------------------------------
// H = X @ W  via V_WMMA_F32_16X16X4_F32 (fp32 matrix core, matches reference)
// One wave -> one 16-row M tile, all 64 output columns (4 accumulators).
// W is staged transposed in LDS so each B fragment is a single ds_load_b64.
// ---------------------------------------------------------------------------
__global__ __launch_bounds__(256) void gcn_gemm64(const float* __restrict__ X,
                                                  const float* __restrict__ W,
                                                  float* __restrict__ H,
                                                  int n_rows) {
    __shared__ float Wt[NODE_D * NODE_D];   // Wt[col][k] = W[k][col]
    for (int i = threadIdx.x; i < NODE_D * NODE_D; i += blockDim.x) {
        int k = i >> 6, c = i & 63;
        Wt[c * NODE_D + k] = W[i];
    }
    __syncthreads();

    const int wave = threadIdx.x >> 5;                 // 0..7
    const int lane = threadIdx.x & 31;
    const int mtile = blockIdx.x * 8 + wave;           // 16 rows per tile
    const int mtiles = (n_rows + 15) >> 4;
    if (mtile >= mtiles) return;                       // wave-uniform exit

    const int m     = lane & 15;                       // row within tile
    const int khalf = (lane >> 4) * 2;                 // 0 or 2
    const int row   = mtile * 16 + m;

    v8f acc0 = {}, acc1 = {}, acc2 = {}, acc3 = {};

    const float* xrow = X + (size_t)row * NODE_D;
#pragma unroll
    for (int k0 = 0; k0 < NODE_D; k0 += 4) {
        // A fragment: lane holds X[row][k0+khalf], X[row][k0+khalf+1]
        v2f a = *(const v2f*)(xrow + k0 + khalf);
        // B fragments per 16-col tile: Wt[col][k0+khalf .. +1] (contiguous)
        v2f b0 = *(const v2f*)(&Wt[(0 * 16 + m) * NODE_D + k0 + khalf]);
        v2f b1 = *(const v2f*)(&Wt[(1 * 16 + m) * NODE_D + k0 + khalf]);
        v2f b2 = *(const v2f*)(&Wt[(2 * 16 + m) * NODE_D + k0 + khalf]);
        v2f b3 = *(const v2f*)(&Wt[(3 * 16 + m) * NODE_D + k0 + khalf]);
        acc0 = __builtin_amdgcn_wmma_f32_16x16x4_f32(false, a, false, b0,
                                                     (short)0, acc0, false, false);
        acc1 = __builtin_amdgcn_wmma_f32_16x16x4_f32(false, a, false, b1,
                                                     (short)0, acc1, false, false);
        acc2 = __builtin_amdgcn_wmma_f32_16x16x4_f32(false, a, false, b2,
                                                     (short)0, acc2, false, false);
        acc3 = __builtin_amdgcn_wmma_f32_16x16x4_f32(false, a, false, b3,
                                                     (short)0, acc3, false, false);
    }

    // C/D layout: VGPR j -> row (j + 8*(lane>=16)), col = lane&15 per 16-tile
    const int rbase = mtile * 16 + ((lane >> 4) << 3);
    const int cl    = lane & 15;
#pragma unroll
    for (int j = 0; j < 8; ++j) {
        float* o = H + (size_t)(rbase + j) * NODE_D + cl;
        o[0]  = acc0[j];
        o[16] = acc1[j];
        o[32] = acc2[j];
        o[48] = acc3[j];
    }
}

// ---------------------------------------------------------------------------
// Edge scatter: one wave32 per edge; lane f handles features f and f+32.
// agg[dst][:] += dinv[src]*dinv[dst] * h[src][:]
// ---------------------------------------------------------------------------
__global__ __launch_bounds__(256) void gcn_scatter(const int* __restrict__ ei,
                                                   int E,
                                                   const float* __restrict__ dinv,
                                                   const float* __restrict__ H,
                                                   float* __restrict__ AGG) {
    int e = blockIdx.x * (blockDim.x >> 5) + (threadIdx.x >> 5);
    if (e >= E) return;                                // wave-uniform
    e = __builtin_amdgcn_readfirstlane(e);             // force SGPR
    const int lane = threadIdx.x & 31;

    const int s = __builtin_amdgcn_readfirstlane(ei[e]);
    const int d = __builtin_amdgcn_readfirstlane(ei[E + e]);
    const float norm = dinv[s] * dinv[d];

    const float* hs = H + (size_t)s * NODE_D;
    float* ad = AGG + (size_t)d * NODE_D;
    const float v0 = hs[lane]      * norm;
    const float v1 = hs[lane + 32] * norm;
    atomicAdd(&ad[lane],      v0);                     // non-returning f32 atomic
    atomicAdd(&ad[lane + 32], v1);
}

// ---------------------------------------------------------------------------
// out = agg + h * invdeg[node] + bias   (+ optional LeakyReLU)
// ---------------------------------------------------------------------------
__global__ void gcn_finalize(const float* __restrict__ AGG,
                             const float* __restrict__ H,
                             const float* __restrict__ invdeg,
                             const float* __restrict__ bias,
                             float* __restrict__ OUT,
                             int n, int do_leaky) {
    int idx = blockIdx.x * blockDim.x + threadIdx.x;
    if (idx >= n * NODE_D) return;
    int i = idx >> 6;
    int f = idx & 63;
    float v = AGG[idx] + H[idx] * invdeg[i] + bias[f];
    if (do_leaky) v = (v >= 0.0f) ? v : NEG_SLOPE * v;
    OUT[idx] = v;
}

// ---------------------------------------------------------------------------
extern "C" void kernel_launch(void* const* d_in, const int* in_sizes, int n_in,
                              void* d_out, int out_size, void* d_ws, size_t ws_size,
                              hipStream_t stream) {
    const float* x  = (const float*)d_in[0];
    const float* W1 = (const float*)d_in[1];
    const float* b1 = (const float*)d_in[2];
    const float* W2 = (const float*)d_in[3];
    const float* b2 = (const float*)d_in[4];
    const float* W3 = (const float*)d_in[5];
    const float* b3 = (const float*)d_in[6];
    const int*   ei = (const int*)d_in[7];

    const int n = in_sizes[0] / NODE_D;    // 100000 nodes
    const int E = in_sizes[7] / 2;         // 1.6M edges

    float* dinv   = (float*)d_ws;                     // n  (deg -> D^-1/2)
    float* invdeg = dinv + n;                         // n
    float* H      = invdeg + n;                       // n*64
    float* AGG    = H + (size_t)n * NODE_D;           // n*64
    float* OUT    = (float*)d_out;                    // n*64 (also L1 intermediate)

    const size_t featBytes = (size_t)n * NODE_D * sizeof(float);

    const int mtiles     = (n + 15) >> 4;
    const dim3 gemmGrid((mtiles + 7) / 8);
    const dim3 gemmBlk(256);
    const dim3 edgeGrid((E + 7) / 8);                 // 8 waves/block, wave/edge
    const dim3 edgeBlk(256);
    const dim3 eltGrid((n * NODE_D + 255) / 256);
    const dim3 eltBlk(256);

    // --- degrees (edge structure is layer-invariant) ---
    hipMemsetAsync(dinv, 0, (size_t)n * sizeof(float), stream);
    gcn_deg_count<<<(E + 255) / 256, 256, 0, stream>>>(ei + E, E, dinv);
    gcn_deg_finalize<<<(n + 255) / 256, 256, 0, stream>>>(dinv, invdeg, n);

    // --- layer 1: conv -> LeakyReLU, result in OUT ---
    gcn_gemm64<<<gemmGrid, gemmBlk, 0, stream>>>(x, W1, H, n);
    hipMemsetAsync(AGG, 0, featBytes, stream);
    gcn_scatter<<<edgeGrid, edgeBlk, 0, stream>>>(ei, E, dinv, H, AGG);
    gcn_finalize<<<eltGrid, eltBlk, 0, stream>>>(AGG, H, invdeg, b1, OUT, n, 1);

    // --- layer 2: conv (no activation), finalize in place -> AGG holds t2 ---
    gcn_gemm64<<<gemmGrid, gemmBlk, 0, stream>>>(OUT, W2, H, n);
    hipMemsetAsync(AGG, 0, featBytes, stream);
    gcn_scatter<<<edgeGrid, edgeBlk, 0, stream>>>(ei, E, dinv, H, AGG);
    gcn_finalize<<<eltGrid, eltBlk, 0, stream>>>(AGG, H, invdeg, b2, AGG, n, 0);

    // --- layer 3: conv -> LeakyReLU, result in OUT ---
    gcn_gemm64<<<gemmGrid, gemmBlk, 0, stream>>>(AGG, W3, H, n);
    hipMemsetAsync(AGG, 0, featBytes, stream);
    gcn_scatter<<<edgeGrid, edgeBlk, 0, stream>>>(ei, E, dinv, H, AGG);
    gcn_finalize<<<eltGrid, eltBlk, 0, stream>>>(AGG, H, invdeg, b3, OUT, n, 1);
}